// LearnableDCT_4449586118806
// MI455X (gfx1250) — compile-verified
//
#include <hip/hip_runtime.h>

typedef float v2f __attribute__((ext_vector_type(2)));
typedef float v4f __attribute__((ext_vector_type(4)));
typedef float v8f __attribute__((ext_vector_type(8)));

// Problem constants (fixed by the reference): x (32,3,512,512) f32, basis (64,64) f32
#define NB    32
#define NC    3
#define HW    512
#define HB    64          // 512/8
#define WB    64
#define ROWS  (NB*NC*HB)  // flattened (bi,ci,hb) rows = 6144
#define XSTR  68          // LDS stride for x tiles  [block][n], conflict-free trick
#define BSTR  68          // LDS stride for basis    [k][n]

__global__ __launch_bounds__(256) void dct_wmma_kernel(
    const float* __restrict__ x,
    const float* __restrict__ basis,
    float* __restrict__ out)
{
    __shared__ float lx[128 * XSTR];  // 128 blocks x 64 pixels (padded)
    __shared__ float lb[64 * BSTR];   // 64x64 basis (padded)

    const int tid     = threadIdx.x;
    const int rowBase = blockIdx.x * 2;   // two (bi,ci,hb) rows per workgroup

    // ---- Stage basis (4096 floats) into LDS, coalesced b128 ----
    #pragma unroll
    for (int u = 0; u < 4; ++u) {
        int idx  = u * 256 + tid;         // float4 index
        int flat = idx * 4;               // = k*64 + col
        int k    = flat >> 6;
        int colb = flat & 63;
        v4f v = *(const v4f*)(basis + flat);
        *(v4f*)(&lb[k * BSTR + colb]) = v;
    }

    // ---- Stage x: 2 block-rows * 8 image-rows * 512 floats = 2048 float4 ----
    #pragma unroll
    for (int u = 0; u < 8; ++u) {
        int idx = u * 256 + tid;          // float4 id
        int rr  = idx >> 10;              // which of the 2 block-rows
        int rem = idx & 1023;
        int i   = rem >> 7;               // image row within 8x8 block (0..7)
        int f4  = rem & 127;              // float4 along W (0..127)
        int r   = rowBase + rr;
        int bi  = r / (NC * HB);
        int r2  = r - bi * (NC * HB);
        int ci  = r2 >> 6;
        int hb  = r2 & 63;
        const float* px = x + (((size_t)(bi * NC + ci) * HW) + (size_t)hb * 8 + i) * HW + f4 * 4;
        v4f v = *(const v4f*)px;
        int wb  = f4 >> 1;
        int j   = (f4 & 1) * 4;
        int n   = i * 8 + j;              // pixel index within block (multiple of 4)
        int blk = rr * 64 + wb;           // local block column 0..127
        *(v4f*)(&lx[blk * XSTR + n]) = v; // 16B-aligned ds_store_b128
    }
    __syncthreads();

    // ---- WMMA GEMM: D(16x16) += A(16x4 of basis) x B(4x16 of blocks) ----
    const int wv   = tid >> 5;            // wave id 0..7
    const int lane = tid & 31;
    const int lcol = lane & 15;
    const int half = lane >> 4;           // 0: lanes 0-15, 1: lanes 16-31
    const int col  = wv * 16 + lcol;      // this lane's block column (N)

    v8f acc0 = {}, acc1 = {}, acc2 = {}, acc3 = {};

    #pragma unroll
    for (int t = 0; t < 16; ++t) {
        const int nb = 4 * t + 2 * half;  // K offset for this half-wave
        // B fragment: xb[nb..nb+1][col]  (conflict-free ds_load_b64)
        v2f b  = *(const v2f*)(&lx[col * XSTR + nb]);
        // A fragments: basis[16m + lcol][nb..nb+1]
        v2f a0 = *(const v2f*)(&lb[( 0 + lcol) * BSTR + nb]);
        v2f a1 = *(const v2f*)(&lb[(16 + lcol) * BSTR + nb]);
        v2f a2 = *(const v2f*)(&lb[(32 + lcol) * BSTR + nb]);
        v2f a3 = *(const v2f*)(&lb[(48 + lcol) * BSTR + nb]);
        acc0 = __builtin_amdgcn_wmma_f32_16x16x4_f32(false, a0, false, b, (short)0, acc0, false, false);
        acc1 = __builtin_amdgcn_wmma_f32_16x16x4_f32(false, a1, false, b, (short)0, acc1, false, false);
        acc2 = __builtin_amdgcn_wmma_f32_16x16x4_f32(false, a2, false, b, (short)0, acc2, false, false);
        acc3 = __builtin_amdgcn_wmma_f32_16x16x4_f32(false, a3, false, b, (short)0, acc3, false, false);
    }

    // ---- Store: out[(bi*C+ci)*64 + k][hb][wb], plane stride 4096 floats ----
    {
        int rrS = col >> 6;
        int wbS = col & 63;
        int r   = rowBase + rrS;
        int bi  = r / (NC * HB);
        int r2  = r - bi * (NC * HB);
        int ci  = r2 >> 6;
        int hb  = r2 & 63;
        float* pout = out + (size_t)((bi * NC + ci) * 64) * 4096 + (size_t)hb * 64 + wbS;
        #pragma unroll
        for (int r8 = 0; r8 < 8; ++r8) {
            int k0 = r8 + 8 * half;       // C/D layout: VGPR r -> M=r (lo half), M=r+8 (hi half)
            pout[(size_t)(k0 +  0) * 4096] = acc0[r8];
            pout[(size_t)(k0 + 16) * 4096] = acc1[r8];
            pout[(size_t)(k0 + 32) * 4096] = acc2[r8];
            pout[(size_t)(k0 + 48) * 4096] = acc3[r8];
        }
    }
}

extern "C" void kernel_launch(void* const* d_in, const int* in_sizes, int n_in,
                              void* d_out, int out_size, void* d_ws, size_t ws_size,
                              hipStream_t stream) {
    const float* x     = (const float*)d_in[0];
    const float* basis = (const float*)d_in[1];
    float* out         = (float*)d_out;
    (void)in_sizes; (void)n_in; (void)out_size; (void)d_ws; (void)ws_size;
    // 6144 flattened (bi,ci,hb) rows, 2 per workgroup -> 3072 WGs, 256 threads (8 wave32)
    dct_wmma_kernel<<<ROWS / 2, 256, 0, stream>>>(x, basis, out);
}